// SpikingNeuralNetwork_50964081934748
// MI455X (gfx1250) — compile-verified
//
#include <hip/hip_runtime.h>

// SNN forward, 100 steps. The reference stores the SPIKE as recurrent state
// (swapped-return bug); spikes are in {0,1} so reset=(state>1) is always 0:
//   m_new = 0.9*s_prev + cur ; s_new = (m_new > 1) ; feed/emit m_new.
//
// cur0 = x@W0^T+b0 once; per step: elementwise LIF0 + two f32 WMMA GEMMs
// (V_WMMA_F32_16X16X4_F32) with fused bias+LIF epilogues. Global->LDS staging
// uses CDNA5 GLOBAL_LOAD_ASYNC_TO_LDS_B128 with ping-pong buffers and a
// single barrier per K-stage (ASYNCcnt synchronization).

typedef float v2f __attribute__((ext_vector_type(2)));
typedef float v8f __attribute__((ext_vector_type(8)));

#define BETA 0.9f
#define THRESH 1.0f
#define NUM_STEPS 100
#define B_ 256
#define D_IN 1024
#define H_ 2048
#define D_OUT 1024

// GEMM tiling: 64x64 block tile, 128 threads (4 waves, 2x2 wave grid,
// each wave = 32x32 tile = 2x2 WMMA accumulators), BK=32.
#define BM 64
#define BN 64
#define BK 32
#define LDSS (BK + 4)        // padded LDS stride (floats), conflict-free
#define TILE_F (BM * LDSS)   // floats per LDS buffer

__global__ __launch_bounds__(256)
void snn_lif0(const float* __restrict__ cur0, float* __restrict__ m0,
              float* __restrict__ s0, int n) {
  int i = blockIdx.x * blockDim.x + threadIdx.x;
  if (i < n) {
    float m = BETA * s0[i] + cur0[i];
    m0[i] = m;
    s0[i] = (m > THRESH) ? 1.0f : 0.0f;
  }
}

// Issue async DMA of one 64x32 f32 tile (row-major, row stride K) into LDS.
// 128 threads, 4 float4 each; 8 consecutive threads cover one 128B row.
__device__ __forceinline__ void stage_tile_async(const float* __restrict__ g,
                                                 long row0, int K, int k0,
                                                 const float* lds, int tid) {
#pragma unroll
  for (int it = 0; it < 4; ++it) {
    int idx = tid + it * 128;
    int r = idx >> 3;
    int c = (idx & 7) * 4;
    const float* gp = g + (row0 + r) * (long)K + (k0 + c);
    unsigned lo = (unsigned)(unsigned long long)(const void*)(lds + r * LDSS + c);
    asm volatile("global_load_async_to_lds_b128 %0, %1, off"
                 :: "v"(lo), "v"(gp) : "memory");
  }
}

__device__ __forceinline__ void wait_async_barrier() {
  asm volatile("s_wait_asynccnt 0x0" ::: "memory");
  __syncthreads();
}

// C[M,N] = A[M,K] @ W[N,K]^T + bias, optionally fused LIF.
// grid = (N/BN, M/BM), block = 128 threads.
template <int DO_LIF>
__global__ __launch_bounds__(128)
void snn_gemm(const float* __restrict__ A, const float* __restrict__ W,
              const float* __restrict__ bias, float* __restrict__ outD,
              float* __restrict__ state, int N, int K) {
  __shared__ float As[2 * TILE_F];
  __shared__ float Bs[2 * TILE_F];

  const int tid  = threadIdx.x;
  const int lane = tid & 31;
  const int l16  = lane & 15;  // N (or M) index within a 16-wide fragment
  const int half = lane >> 4;  // half-wave select
  const int wave = tid >> 5;
  const int wm   = (wave & 1) * 32;  // wave row offset in block tile
  const int wn   = (wave >> 1) * 32; // wave col offset in block tile
  const long bm0 = (long)blockIdx.y * BM;
  const long bn0 = (long)blockIdx.x * BN;

  v8f acc[2][2] = {};

  const int S = K / BK;

  // Prologue: stage 0 into buffer 0.
  stage_tile_async(A, bm0, K, 0, As, tid);
  stage_tile_async(W, bn0, K, 0, Bs, tid);
  wait_async_barrier();

  for (int s = 0; s < S; ++s) {
    const float* curA = As + (s & 1) * TILE_F;
    const float* curB = Bs + (s & 1) * TILE_F;

    // Prefetch next stage into the other buffer (DMA overlaps WMMA below).
    if (s + 1 < S) {
      const float* nA = As + ((s + 1) & 1) * TILE_F;
      const float* nB = Bs + ((s + 1) & 1) * TILE_F;
      stage_tile_async(A, bm0, K, (s + 1) * BK, nA, tid);
      stage_tile_async(W, bn0, K, (s + 1) * BK, nB, tid);
    }

#pragma unroll
    for (int kk = 0; kk < BK / 4; ++kk) {
      // ISA 32-bit A layout: lanes 0-15 hold K={0,1}, lanes 16-31 K={2,3}
      const int kb = kk * 4 + half * 2;
      v2f af[2], bf[2];
#pragma unroll
      for (int i = 0; i < 2; ++i) {
        const float* p = &curA[(wm + i * 16 + l16) * LDSS + kb];
        af[i][0] = p[0];
        af[i][1] = p[1];
      }
#pragma unroll
      for (int j = 0; j < 2; ++j) {
        const float* p = &curB[(wn + j * 16 + l16) * LDSS + kb];
        bf[j][0] = p[0];
        bf[j][1] = p[1];
      }
#pragma unroll
      for (int i = 0; i < 2; ++i)
#pragma unroll
        for (int j = 0; j < 2; ++j)
          acc[i][j] = __builtin_amdgcn_wmma_f32_16x16x4_f32(
              false, af[i], false, bf[j], (short)0, acc[i][j], false, false);
    }

    // Single barrier per stage: next-stage DMA done + all waves done reading.
    wait_async_barrier();
  }

  // Epilogue: C/D layout -> lane l16 = N, VGPR r = M row (half adds +8)
#pragma unroll
  for (int j = 0; j < 2; ++j) {
    const long col = bn0 + wn + j * 16 + l16;
    const float bv = bias[col];
#pragma unroll
    for (int i = 0; i < 2; ++i) {
#pragma unroll
      for (int r = 0; r < 8; ++r) {
        const long row = bm0 + wm + i * 16 + half * 8 + r;
        const long off = row * (long)N + col;
        const float cur = acc[i][j][r] + bv;
        if (DO_LIF) {
          const float m = BETA * state[off] + cur;
          outD[off]  = m;
          state[off] = (m > THRESH) ? 1.0f : 0.0f;
        } else {
          outD[off] = cur;
        }
      }
    }
  }
}

extern "C" void kernel_launch(void* const* d_in, const int* in_sizes, int n_in,
                              void* d_out, int out_size, void* d_ws, size_t ws_size,
                              hipStream_t stream) {
  const float* x  = (const float*)d_in[0]; // [256,1024]
  const float* W0 = (const float*)d_in[1]; // [2048,1024]
  const float* b0 = (const float*)d_in[2]; // [2048]
  const float* W1 = (const float*)d_in[3]; // [2048,2048]
  const float* b1 = (const float*)d_in[4]; // [2048]
  const float* W2 = (const float*)d_in[5]; // [1024,2048]
  const float* b2 = (const float*)d_in[6]; // [1024]
  float* out = (float*)d_out;              // [100,256,1024]

  // Workspace layout (floats)
  float* ws   = (float*)d_ws;
  float* cur0 = ws;                        // 256*2048
  float* m0   = cur0 + (size_t)B_ * H_;    // 256*2048
  float* m1   = m0 + (size_t)B_ * H_;      // 256*2048
  float* s0   = m1 + (size_t)B_ * H_;      // 256*2048
  float* s1   = s0 + (size_t)B_ * H_;      // 256*2048
  float* s2   = s1 + (size_t)B_ * H_;      // 256*1024

  // Zero the recurrent spike states (s0|s1|s2 are contiguous)
  hipMemsetAsync(s0, 0,
                 ((size_t)2 * B_ * H_ + (size_t)B_ * D_OUT) * sizeof(float),
                 stream);

  // cur0 = x @ W0^T + b0 (hoisted, no LIF)
  snn_gemm<0><<<dim3(H_ / BN, B_ / BM), dim3(128), 0, stream>>>(
      x, W0, b0, cur0, nullptr, H_, D_IN);

  for (int t = 0; t < NUM_STEPS; ++t) {
    snn_lif0<<<(B_ * H_) / 256, dim3(256), 0, stream>>>(cur0, m0, s0, B_ * H_);
    snn_gemm<1><<<dim3(H_ / BN, B_ / BM), dim3(128), 0, stream>>>(
        m0, W1, b1, m1, s1, H_, H_);
    snn_gemm<1><<<dim3(D_OUT / BN, B_ / BM), dim3(128), 0, stream>>>(
        m1, W2, b2, out + (size_t)t * B_ * D_OUT, s2, D_OUT, H_);
  }
}